// Encoder_6107443495308
// MI455X (gfx1250) — compile-verified
//
#include <hip/hip_runtime.h>

typedef __attribute__((ext_vector_type(2))) float v2f;
typedef __attribute__((ext_vector_type(8))) float v8f;

static constexpr int CH = 64; // hidden == out channels

__device__ __forceinline__ void atomic_add_f32(float* p, float v) {
#if defined(__HIP_DEVICE_COMPILE__)
    unsafeAtomicAdd(p, v);   // lowers to global_atomic_add_f32
#else
    atomicAdd(p, v);
#endif
}

// ---------------- degree / normalization ----------------

__global__ void k_deg_init(float* deg, int n) {
    int i = blockIdx.x * blockDim.x + threadIdx.x;
    if (i < n) deg[i] = 1.0f;  // self-loop
}

__global__ void k_deg_count(const long long* __restrict__ ei, long long E,
                            float* __restrict__ deg) {
    long long stride = (long long)gridDim.x * blockDim.x;
    for (long long e = blockIdx.x * (long long)blockDim.x + threadIdx.x; e < E; e += stride) {
        int dst = (int)ei[E + e];
        atomic_add_f32(&deg[dst], 1.0f);
    }
}

__global__ void k_deg_rsqrt(float* deg, int n) {
    int i = blockIdx.x * blockDim.x + threadIdx.x;
    if (i < n) {
        float d = deg[i];
        deg[i] = d > 0.0f ? rsqrtf(d) : 0.0f;
    }
}

__global__ void k_zero(float* __restrict__ p, long long n) {
    long long stride = (long long)gridDim.x * blockDim.x;
    for (long long i = blockIdx.x * (long long)blockDim.x + threadIdx.x; i < n; i += stride)
        p[i] = 0.0f;
}

// ---------------- WMMA GEMM:  out[r,n] = dinv[r] * sum_k A[r,k]*B[k,n] ----------------
// A: [M,K] row-major, K in {256,64} (power of 2, kshift = log2 K). B: [K,64] row-major.
// Block: 128 threads = 4 waves; block computes rows [row0, row0+16), each wave one
// 16x16 N-tile. A panel staged in LDS with padded stride (bank-conflict free).

__global__ __launch_bounds__(128) void k_gemm_wmma(const float* __restrict__ A,
                                                   const float* __restrict__ B,
                                                   const float* __restrict__ dinv,
                                                   float* __restrict__ out,
                                                   int M, int K, int kshift) {
    __shared__ float As[16 * (256 + 4)];        // max K = 256, stride K+4
    const int lds_stride = K + 4;
    const int row0 = blockIdx.x * 16;
    const int tid  = threadIdx.x;
    const int kmask = K - 1;

    // cooperative, coalesced A-panel load (zero-pad rows beyond M)
    for (int idx = tid; idx < (16 << kshift); idx += 128) {
        int r = idx >> kshift;
        int k = idx & kmask;
        int gr = row0 + r;
        As[r * lds_stride + k] = (gr < M) ? A[(size_t)gr * K + k] : 0.0f;
    }
    __syncthreads();

    const int lane = tid & 31;
    const int wave = tid >> 5;
    const int n    = lane & 15;       // N index (also M index for A fragment)
    const int half = lane >> 4;       // 0: lanes 0-15, 1: lanes 16-31
    const int n0   = wave * 16;       // this wave's N-tile origin

    v8f c = {};
    for (int k = 0; k < K; k += 4) {
        const int kk = k + 2 * half;
        // A fragment: lane -> row M=n ; VGPR0/1 -> K = kk, kk+1
        v2f a;
        a.x = As[n * lds_stride + kk];
        a.y = As[n * lds_stride + kk + 1];
        // B fragment: VGPR v, half h -> K = k + v + 2h ; lane -> N
        v2f b;
        b.x = B[(size_t)kk       * CH + n0 + n];
        b.y = B[(size_t)(kk + 1) * CH + n0 + n];
        c = __builtin_amdgcn_wmma_f32_16x16x4_f32(false, a, false, b,
                                                  (short)0, c, false, false);
    }

    // D layout: VGPR v -> row M = v + 8*half, lane -> N. Fuse dinv row-scale.
    #pragma unroll
    for (int v = 0; v < 8; ++v) {
        int mg = row0 + v + 8 * half;
        if (mg < M)
            out[(size_t)mg * CH + n0 + n] = c[v] * dinv[mg];
    }
}

// ---------------- edge scatter:  agg[dst] += hs[src]  (64 ch, one wave per edge) ------

__global__ void k_scatter(const long long* __restrict__ ei, long long E,
                          const float* __restrict__ hs, float* __restrict__ agg) {
    long long nwork = E << 5;                   // 32 lanes per edge, float2 per lane
    long long stride = (long long)gridDim.x * blockDim.x;
    for (long long i = blockIdx.x * (long long)blockDim.x + threadIdx.x; i < nwork; i += stride) {
        long long e = i >> 5;
        int c0 = ((int)i & 31) * 2;
        int src = (int)ei[e];                   // lanes of a wave share e -> scalar load
        int dst = (int)ei[E + e];
        const float2 v = *(const float2*)(hs + (size_t)src * CH + c0);
        float* d = agg + (size_t)dst * CH + c0;
        atomic_add_f32(d,     v.x);
        atomic_add_f32(d + 1, v.y);
    }
}

// ---------------- epilogue:  io = [relu]( dinv[node]*(io + hs) + bias[ch] ) ----------

__global__ void k_post(float* __restrict__ io, const float* __restrict__ hs,
                       const float* __restrict__ dinv, const float* __restrict__ bias,
                       long long total, int relu) {
    long long stride = (long long)gridDim.x * blockDim.x;
    for (long long i = blockIdx.x * (long long)blockDim.x + threadIdx.x; i < total; i += stride) {
        int node = (int)(i >> 6);
        int ch   = (int)i & 63;
        float v = dinv[node] * (io[i] + hs[i]) + bias[ch];
        if (relu) v = fmaxf(v, 0.0f);
        io[i] = v;
    }
}

// ---------------------------------------------------------------------------

extern "C" void kernel_launch(void* const* d_in, const int* in_sizes, int n_in,
                              void* d_out, int out_size, void* d_ws, size_t ws_size,
                              hipStream_t stream) {
    const float*     x  = (const float*)d_in[0];
    const long long* ei = (const long long*)d_in[1];   // int64 edge_index [2,E]
    const float*     W1 = (const float*)d_in[2];
    const float*     b1 = (const float*)d_in[3];
    const float*     W2 = (const float*)d_in[4];
    const float*     b2 = (const float*)d_in[5];
    float*           out = (float*)d_out;

    const int       IN_CH = 256;
    const int       N = in_sizes[0] / IN_CH;           // 100000
    const long long E = in_sizes[1] / 2;               // 3200000
    const long long NF = (long long)N * CH;

    char* ws = (char*)d_ws;
    float* dinv = (float*)ws;                          // N floats
    size_t off  = ((size_t)N * sizeof(float) + 255) & ~(size_t)255;
    float* hs   = (float*)(ws + off);                  // N*64: scaled GEMM output
    float* y1   = (float*)(ws + off + (size_t)NF * sizeof(float)); // N*64: layer-1 out

    const int T = 256;
    dim3 gN((N + T - 1) / T);
    dim3 gF(4096);
    dim3 gE(8192);
    dim3 gG((N + 15) / 16);

    // normalization: deg -> dinv (in place)
    k_deg_init <<<gN, T, 0, stream>>>(dinv, N);
    k_deg_count<<<dim3(2048), T, 0, stream>>>(ei, E, dinv);
    k_deg_rsqrt<<<gN, T, 0, stream>>>(dinv, N);

    // ---- layer 1:  y1 = relu( D^-1/2 A D^-1/2 (x W1) + b1 )
    k_gemm_wmma<<<gG, 128, 0, stream>>>(x, W1, dinv, hs, N, 256, 8);
    k_zero     <<<gF, T, 0, stream>>>(y1, NF);
    k_scatter  <<<gE, T, 0, stream>>>(ei, E, hs, y1);
    k_post     <<<gF, T, 0, stream>>>(y1, hs, dinv, b1, NF, 1);

    // ---- layer 2:  out = D^-1/2 A D^-1/2 (y1 W2) + b2   (accumulate in d_out)
    k_gemm_wmma<<<gG, 128, 0, stream>>>(y1, W2, dinv, hs, N, 64, 6);
    k_zero     <<<gF, T, 0, stream>>>(out, NF);
    k_scatter  <<<gE, T, 0, stream>>>(ei, E, hs, out);
    k_post     <<<gF, T, 0, stream>>>(out, hs, dinv, b2, NF, 0);
}